// VQEmbedding_76742475645286
// MI455X (gfx1250) — compile-verified
//
#include <hip/hip_runtime.h>
#include <hip/hip_bf16.h>

// VQ-VAE nearest-codebook lookup for MI455X (gfx1250, wave32, WMMA).
//
// argmin_k ( ||e_k||^2 - 2 * z . e_k )  -- ||z||^2 dropped (constant per row).
//
// v2: codebook staged into LDS via CDNA5 async-to-LDS copies (ASYNCcnt),
//     code norms precomputed once per workgroup, M-blocked x2 per wave
//     (32 fp32 WMMAs per B-tile), B sourced from LDS (ds_load_b64),
//     non-temporal z loads / index stores.

typedef __attribute__((ext_vector_type(2))) float v2f;
typedef __attribute__((ext_vector_type(4))) float v4f;
typedef __attribute__((ext_vector_type(8))) float v8f;

#define D_DIM    64
#define HW_SZ    4096                   // H*W = 64*64 (multiple of 32)
#define K_CODES  512
#define K_TILES  (K_CODES / 16)
#define SMEM_BYTES ((K_CODES * D_DIM + K_CODES) * 4)   // 128KB weights + 2KB norms

__global__ __launch_bounds__(256) void
vq_argmin_wmma_f32(const float* __restrict__ z,     // (B, D=64, H, W)
                   const float* __restrict__ wt,    // (512, 64) row-major
                   float* __restrict__ out)         // (B*H*W) indices as float
{
    extern __shared__ float smem[];
    float* sW = smem;                      // 512 x 64 codebook
    float* sN = smem + K_CODES * D_DIM;    // 512 code norms

    const int tid  = threadIdx.x;
    const int lane = tid & 31;
    const int wave = tid >> 5;
    const int n    = lane & 15;            // code column / A row within tile
    const int hi   = lane >> 4;            // half-wave select

    // ---- Stage codebook into LDS: async global->LDS B128 copies ---------
    {
        const uint32_t lds_base = (uint32_t)(uintptr_t)sW;   // low 32 = LDS offset
        for (int i = tid * 4; i < K_CODES * D_DIM; i += 256 * 4) {
            const float* gp   = wt + i;
            const uint32_t lo = lds_base + (uint32_t)i * 4u;
            asm volatile("global_load_async_to_lds_b128 %0, %1, off"
                         :: "v"(lo), "v"(gp) : "memory");
        }
        asm volatile("s_wait_asynccnt 0" ::: "memory");
    }
    __syncthreads();

    // ---- Precompute ||e_k||^2 once per workgroup (from LDS) -------------
    for (int c = tid; c < K_CODES; c += 256) {
        const float* p = sW + c * D_DIM;
        float s = 0.0f;
#pragma unroll
        for (int d = 0; d < D_DIM; d += 4) {
            v4f v = *(const v4f*)(p + d);
            s = fmaf(v.x, v.x, s); s = fmaf(v.y, v.y, s);
            s = fmaf(v.z, v.z, s); s = fmaf(v.w, v.w, s);
        }
        sN[c] = s;
    }
    __syncthreads();

    // ---- Load A: two 16-row subtiles (32 rows) per wave -----------------
    // ISA 16x4 fp32 A layout: a[2t]   = K=4t   (lanes 0-15) / K=4t+2 (lanes 16-31)
    //                         a[2t+1] = K=4t+1 (lanes 0-15) / K=4t+3 (lanes 16-31)
    const int rowBase = blockIdx.x * 256 + wave * 32;   // never crosses b (HW%32==0)
    const int bimg = rowBase / HW_SZ;
    const int s0   = rowBase % HW_SZ;
    const float* za = z + (size_t)bimg * (D_DIM * HW_SZ) + (s0 + n);

    float a0[32], a1[32];
#pragma unroll
    for (int t = 0; t < 16; ++t) {
        const int c0 = 4 * t + 2 * hi;
        a0[2*t]   = __builtin_nontemporal_load(za + (size_t)c0       * HW_SZ);
        a0[2*t+1] = __builtin_nontemporal_load(za + (size_t)(c0 + 1) * HW_SZ);
        a1[2*t]   = __builtin_nontemporal_load(za + (size_t)c0       * HW_SZ + 16);
        a1[2*t+1] = __builtin_nontemporal_load(za + (size_t)(c0 + 1) * HW_SZ + 16);
    }

    float bestv0[8], bestv1[8];
    int   besti0[8], besti1[8];
#pragma unroll
    for (int j = 0; j < 8; ++j) {
        bestv0[j] = __builtin_huge_valf(); besti0[j] = 0;
        bestv1[j] = __builtin_huge_valf(); besti1[j] = 0;
    }

    // Per-lane base into LDS codebook: code n, feature pair (2*hi, 2*hi+1) mod 4.
    const float* wb = sW + n * D_DIM + 2 * hi;
    const float* nb = sN + n;

#pragma unroll 1
    for (int tt = 0; tt < K_TILES; ++tt) {
        // ---- B tile (16 codes x 64 feats) from LDS: 16x ds_load_b64 ----
        const float* wp = wb + tt * 16 * D_DIM;
        float bm[32];
#pragma unroll
        for (int t = 0; t < 16; ++t) {
            v2f v = *(const v2f*)(wp + 4 * t);
            bm[2*t] = v.x; bm[2*t+1] = v.y;
        }
        const float nrm = nb[tt * 16];   // ||e_{n0+n}||^2 (precomputed)

        // ---- 2 x 16 chained fp32 WMMAs sharing the B tile --------------
        v8f c0 = {}, c1 = {};
#pragma unroll
        for (int t = 0; t < 16; ++t) {
            v2f bv;  bv.x  = bm[2*t]; bv.y  = bm[2*t+1];
            v2f av0; av0.x = a0[2*t]; av0.y = a0[2*t+1];
            v2f av1; av1.x = a1[2*t]; av1.y = a1[2*t+1];
            c0 = __builtin_amdgcn_wmma_f32_16x16x4_f32(
                     false, av0, false, bv, (short)0, c0, false, false);
            c1 = __builtin_amdgcn_wmma_f32_16x16x4_f32(
                     false, av1, false, bv, (short)0, c1, false, false);
        }

        // ---- Fused distance + running argmin (first-min tie-break) -----
        const int kidx = tt * 16 + n;
#pragma unroll
        for (int j = 0; j < 8; ++j) {
            const float d0 = fmaf(-2.0f, c0[j], nrm);
            const float d1 = fmaf(-2.0f, c1[j], nrm);
            if (d0 < bestv0[j]) { bestv0[j] = d0; besti0[j] = kidx; }
            if (d1 < bestv1[j]) { bestv1[j] = d1; besti1[j] = kidx; }
        }
    }

    // ---- Cross-lane argmin within each 16-lane group --------------------
    // C layout: VGPR j -> row M=j (lanes 0-15), row M=j+8 (lanes 16-31).
#pragma unroll
    for (int j = 0; j < 8; ++j) {
        float bv = bestv0[j]; int bi = besti0[j];
#pragma unroll
        for (int m = 1; m <= 8; m <<= 1) {
            const float ov = __shfl_xor(bv, m, 32);
            const int   oi = __shfl_xor(bi, m, 32);
            if (ov < bv || (ov == bv && oi < bi)) { bv = ov; bi = oi; }
        }
        besti0[j] = bi;

        bv = bestv1[j]; bi = besti1[j];
#pragma unroll
        for (int m = 1; m <= 8; m <<= 1) {
            const float ov = __shfl_xor(bv, m, 32);
            const int   oi = __shfl_xor(bi, m, 32);
            if (ov < bv || (ov == bv && oi < bi)) { bv = ov; bi = oi; }
        }
        besti1[j] = bi;
    }

    if (n == 0) {          // lanes 0 (rows j) and 16 (rows j+8) of each subtile
#pragma unroll
        for (int j = 0; j < 8; ++j) {
            __builtin_nontemporal_store((float)besti0[j], out + rowBase +      hi * 8 + j);
            __builtin_nontemporal_store((float)besti1[j], out + rowBase + 16 + hi * 8 + j);
        }
    }
}

extern "C" void kernel_launch(void* const* d_in, const int* in_sizes, int n_in,
                              void* d_out, int out_size, void* d_ws, size_t ws_size,
                              hipStream_t stream) {
    const float* z  = (const float*)d_in[0];   // (32, 64, 64, 64) fp32
    const float* wt = (const float*)d_in[1];   // (512, 64) fp32
    float* out = (float*)d_out;                // 131072 indices (as float)

    const int N = in_sizes[0] / D_DIM;         // 131072 rows
    const int blocks = N / 256;                // 8 waves x 32 rows per block

    vq_argmin_wmma_f32<<<blocks, 256, SMEM_BYTES, stream>>>(z, wt, out);
}